// ForwardCASSI_34007551050254
// MI455X (gfx1250) — compile-verified
//
#include <hip/hip_runtime.h>
#include <stdint.h>

// ---------------------------------------------------------------------------
// CASSI forward model for MI455X (gfx1250).
// Memory-bound (260 MB of X @ 23.3 TB/s => ~11-20 us floor). No matmul
// structure => WMMA not applicable; the CDNA5 win is the async global->LDS
// DMA path (ASYNCcnt) for the contiguous per-row staging, wave32 shuffles,
// and the split wait counters.
// ---------------------------------------------------------------------------

#define LBANDS   31
#define MD       256
#define IN_W     512
#define ROWF     (IN_W * LBANDS)      // 15872 floats per input row
#define AUXN     (MD * LBANDS)        // 7936 aux elements per output row
#define YW       (MD + LBANDS - 1)    // 286
#define YN       (8 * MD * YW)        // 585728 output Y elements
#define HN       (MD * MD)            // 65536  output H elements
#define NBLK     512                  // 16 wave32 per block

struct SharedMem {
    float raw0[ROWF];   // core row 2r   -> overwritten in place with V (vertical filter)
    float raw1[ROWF];   // core row 2r+1 -> overlaid with aux (first AUXN floats)
    float hrow[MD];
    float red[16];
};

__global__ __launch_bounds__(NBLK)
void cassi_fwd_kernel(const float* __restrict__ X,
                      const float* __restrict__ H,
                      float* __restrict__ Y,
                      int* __restrict__ gmax) {
    __shared__ SharedMem sm;
    const int tid = threadIdx.x;
    const int blk = blockIdx.x;       // blk = b*256 + r
    const int b   = blk >> 8;
    const int r   = blk & 255;

    // src -> input row 2r of batch b; rows 2r and 2r+1 are one contiguous
    // 2*ROWF*4 = 126976-byte block, 16B-aligned.
    const float* src = X + (((size_t)b * IN_W + 2 * r) * (size_t)IN_W) * LBANDS;

    // ---- Phase 0: async DMA of both core rows into LDS (B128 per lane) ----
    {
        const uint32_t lds_base = (uint32_t)(uintptr_t)(&sm.raw0[0]); // flat LDS addr low bits == LDS offset
        for (int k = tid; k < (2 * ROWF) / 4; k += NBLK) {            // 7936 16-byte chunks, exact
            uint32_t lds_off = lds_base + (uint32_t)k * 16u;
            const float* g = src + (size_t)k * 4;
            asm volatile("global_load_async_to_lds_b128 %0, %1, off"
                         :: "v"(lds_off), "v"(g) : "memory");
        }
    }
    // Mask row through the normal vector path (tiny, cache-resident).
    if (tid < MD) sm.hrow[tid] = H[r * MD + tid];

    asm volatile("s_wait_asynccnt 0x0" ::: "memory");   // our wave's DMA done
    __syncthreads();                                    // all waves' LDS writes visible

    // ---- Phase A: vertical antialias filter (taps 2r-1,2r,2r+1,2r+2) ------
    // jax.image.resize(bilinear, antialias=True): interior [1/8,3/8,3/8,1/8],
    // edges renormalized to [3/7,3/7,1/7].
    {
        const bool has_m1 = (r > 0);
        const bool has_p2 = (r < MD - 1);
        const float wc = (has_m1 && has_p2) ? 0.375f : (3.0f / 7.0f);
        const float we = (has_m1 && has_p2) ? 0.125f : (1.0f / 7.0f);
        const float* rowm1 = src - ROWF;        // row 2r-1 (neighbor's core: L2 hit)
        const float* rowp2 = src + 2 * ROWF;    // row 2r+2 (neighbor's core: L2 hit)
        for (int k = tid; k < ROWF; k += NBLK) {
            float v = wc * (sm.raw0[k] + sm.raw1[k]);
            if (has_m1) v += we * rowm1[k];
            if (has_p2) v += we * rowp2[k];
            sm.raw0[k] = v;                     // in place: index k owned by one thread
        }
    }
    __syncthreads();

    // ---- Phase B: horizontal antialias filter + coded aperture ------------
    {
        const float* V = sm.raw0;
        float* aux = sm.raw1;
        for (int k = tid; k < AUXN; k += NBLK) {
            const int c = k / LBANDS;
            const int i = k - c * LBANDS;
            const bool hm = (c > 0);
            const bool hp = (c < MD - 1);
            const float wc = (hm && hp) ? 0.375f : (3.0f / 7.0f);
            const float we = (hm && hp) ? 0.125f : (1.0f / 7.0f);
            float v = wc * (V[(2 * c) * LBANDS + i] + V[(2 * c + 1) * LBANDS + i]);
            if (hm) v += we * V[(2 * c - 1) * LBANDS + i];
            if (hp) v += we * V[(2 * c + 2) * LBANDS + i];
            aux[k] = sm.hrow[c] * v;
        }
    }
    __syncthreads();

    // ---- Phase C: spectral shear-accumulate + block max -------------------
    float y = 0.0f;
    if (tid < YW) {
        const int c = tid;
        const int ilo = (c - (MD - 1) > 0) ? (c - (MD - 1)) : 0;
        const int ihi = (c < LBANDS - 1) ? c : (LBANDS - 1);
        const float* aux = sm.raw1;
        // stride of 31 words is coprime with 64 LDS banks -> conflict-free
        for (int i = ilo; i <= ihi; ++i)
            y += aux[(c - i) * LBANDS + i];
        Y[(size_t)blk * YW + c] = y;            // unscaled; finalize divides
    }
    // wave32 max reduction, then cross-wave via LDS
    float m = y;
    #pragma unroll
    for (int off = 16; off > 0; off >>= 1)
        m = fmaxf(m, __shfl_xor(m, off, 32));
    const int wid  = tid >> 5;
    const int lane = tid & 31;
    if (lane == 0) sm.red[wid] = m;
    __syncthreads();
    if (wid == 0) {
        float mm = (lane < 16) ? sm.red[lane] : 0.0f;
        #pragma unroll
        for (int off = 16; off > 0; off >>= 1)
            mm = fmaxf(mm, __shfl_xor(mm, off, 32));
        if (lane == 0)
            atomicMax(gmax, __float_as_int(mm)); // Y >= 0: int order == float order
    }
}

__global__ void cassi_init_kernel(int* gmax) {
    if (threadIdx.x == 0 && blockIdx.x == 0) *gmax = 0;
}

__global__ void cassi_finalize_kernel(float* __restrict__ out,
                                      const float* __restrict__ H,
                                      const int* __restrict__ gmax) {
    const int idx = blockIdx.x * blockDim.x + threadIdx.x;
    const float mx = __int_as_float(*gmax);
    if (idx < YN) {
        out[idx] = out[idx] / mx;               // exact match of Y / max(Y)
    } else if (idx < YN + HN) {
        out[idx] = H[idx - YN];                 // second tuple element: H
    }
}

extern "C" void kernel_launch(void* const* d_in, const int* in_sizes, int n_in,
                              void* d_out, int out_size, void* d_ws, size_t ws_size,
                              hipStream_t stream) {
    (void)in_sizes; (void)n_in; (void)out_size; (void)ws_size;
    const float* X = (const float*)d_in[0];   // (8,512,512,31) f32
    const float* H = (const float*)d_in[1];   // (1,256,256,1,1) f32
    float* out = (float*)d_out;               // Y (585728) ++ H (65536)
    int* gmax = (int*)d_ws;                   // global-max slot (f32 bits)

    cassi_init_kernel<<<1, 64, 0, stream>>>(gmax);
    cassi_fwd_kernel<<<8 * MD, NBLK, 0, stream>>>(X, H, out, gmax);
    const int total = YN + HN;
    cassi_finalize_kernel<<<(total + 255) / 256, 256, 0, stream>>>(out, H, gmax);
}